// InteractionMap_10634339025380
// MI455X (gfx1250) — compile-verified
//
#include <hip/hip_runtime.h>
#include <math.h>

typedef __attribute__((ext_vector_type(2))) float v2f;
typedef __attribute__((ext_vector_type(8))) float v8f;

#define L 800
#define DD 100
#define NIN 1024
#define HH 50
#define K2 200          // 2*D
#define WST 200         // LDS stride for W1 tile (even -> b64-aligned)
#define EST 102         // LDS stride for e rows (even)
#define PIX (L * L)     // 640000
#define PY 89           // pooled size

__device__ __forceinline__ float eluf(float x) {
    return x > 0.0f ? x : expm1f(x);
}

// ---------------------------------------------------------------------------
// Kernel 1: e = elu(z @ W_emb^T + b_emb), for z0 and z1 (blockIdx.y selects)
// ---------------------------------------------------------------------------
__global__ __launch_bounds__(256) void emb_kernel(
    const float* __restrict__ z0, const float* __restrict__ z1,
    const float* __restrict__ W_emb, const float* __restrict__ b_emb,
    float* __restrict__ e0, float* __restrict__ e1)
{
    int idx = blockIdx.x * 256 + threadIdx.x;
    if (idx >= L * DD) return;
    const float* z = blockIdx.y ? z1 : z0;
    float*       e = blockIdx.y ? e1 : e0;
    int l = idx / DD, d = idx % DD;
    const float* zr = z + (size_t)l * NIN;
    const float* wr = W_emb + (size_t)d * NIN;
    float s = 0.0f;
    #pragma unroll 8
    for (int k = 0; k < NIN; ++k) s = fmaf(zr[k], wr[k], s);
    s += b_emb[d];
    e[idx] = eluf(s);
}

// ---------------------------------------------------------------------------
// Kernel 2: Bmap[h,i,j] = elu(bn1(elu( sum_k W1[h,k]*feat_ij[k] + b1[h] )))
// feat_ij[k] = |e0[i,k]-e1[j,k]| (k<100) ; e0[i,k-100]*e1[j,k-100] (k>=100)
// WMMA f32 16x16x4: M=h (4 tiles, 50 padded->64), N=j (16), K=200 (50 steps).
// Block = 16x16 (i,j) tile, 16 waves; wave w owns row i=i0+w.
// ---------------------------------------------------------------------------
__global__ __launch_bounds__(512) void pair_gemm_kernel(
    const float* __restrict__ e0, const float* __restrict__ e1,
    const float* __restrict__ W1, const float* __restrict__ b1,
    const float* __restrict__ bn_g, const float* __restrict__ bn_b,
    const float* __restrict__ bn_m, const float* __restrict__ bn_v,
    float* __restrict__ Bmap)
{
    __shared__ float Wlds[64 * WST];   // [h][k], rows 50..63 zero
    __shared__ float e0l[16 * EST];
    __shared__ float e1l[16 * EST];
    __shared__ float sc[64], sh[64], bias[64];

    const int tid  = threadIdx.x;      // 0..511
    const int wave = tid >> 5;         // 0..15  -> local i
    const int lane = tid & 31;
    const int i0 = blockIdx.x * 16;
    const int j0 = blockIdx.y * 16;

    for (int idx = tid; idx < 64 * WST; idx += 512) Wlds[idx] = 0.0f;
    if (tid < 64) {
        if (tid < HH) {
            float s = bn_g[tid] / sqrtf(bn_v[tid] + 1e-5f);
            sc[tid] = s;
            sh[tid] = bn_b[tid] - bn_m[tid] * s;
            bias[tid] = b1[tid];
        } else { sc[tid] = 0.0f; sh[tid] = 0.0f; bias[tid] = 0.0f; }
    }
    __syncthreads();
    for (int idx = tid; idx < HH * K2; idx += 512) {
        int h = idx / K2, k = idx % K2;
        Wlds[h * WST + k] = W1[idx];
    }
    for (int idx = tid; idx < 16 * DD; idx += 512) {
        int r = idx / DD, k = idx % DD;
        e0l[r * EST + k] = e0[(size_t)(i0 + r) * DD + k];
        e1l[r * EST + k] = e1[(size_t)(j0 + r) * DD + k];
    }
    __syncthreads();

    const int jl    = lane & 15;            // N index (j) and A's M index (h)
    const int khalf = (lane >> 4) << 1;     // 0 or 2: this lane's K sub-offset
    const float* e0row = &e0l[wave * EST];
    const float* e1row = &e1l[jl * EST];

    v8f acc[4] = {v8f{}, v8f{}, v8f{}, v8f{}};

    // K-steps 0..24: abs-difference features (k = 0..99)
    #pragma unroll 2
    for (int ks = 0; ks < 25; ++ks) {
        const int kb = ks * 4 + khalf;
        v2f a = *(const v2f*)&e0row[kb];
        v2f c = *(const v2f*)&e1row[kb];
        v2f bfrag;
        bfrag.x = fabsf(a.x - c.x);
        bfrag.y = fabsf(a.y - c.y);
        #pragma unroll
        for (int t = 0; t < 4; ++t) {
            v2f af = *(const v2f*)&Wlds[(t * 16 + jl) * WST + kb];
            acc[t] = __builtin_amdgcn_wmma_f32_16x16x4_f32(
                false, af, false, bfrag, (short)0, acc[t], false, false);
        }
    }
    // K-steps 25..49: product features (k = 100..199)
    #pragma unroll 2
    for (int ks = 25; ks < 50; ++ks) {
        const int kb = ks * 4 + khalf;     // 100..199 (index into W1 K dim)
        const int kk = kb - 100;           // index into e rows
        v2f a = *(const v2f*)&e0row[kk];
        v2f c = *(const v2f*)&e1row[kk];
        v2f bfrag;
        bfrag.x = a.x * c.x;
        bfrag.y = a.y * c.y;
        #pragma unroll
        for (int t = 0; t < 4; ++t) {
            v2f af = *(const v2f*)&Wlds[(t * 16 + jl) * WST + kb];
            acc[t] = __builtin_amdgcn_wmma_f32_16x16x4_f32(
                false, af, false, bfrag, (short)0, acc[t], false, false);
        }
    }

    // Epilogue: C layout -> lane l, vgpr r: M = r + 8*(l>=16), N = l&15
    const int  mofs  = (lane & 16) ? 8 : 0;
    const int  i     = i0 + wave;
    const size_t pb  = (size_t)i * L + (size_t)(j0 + jl);
    #pragma unroll
    for (int t = 0; t < 4; ++t) {
        #pragma unroll
        for (int r = 0; r < 8; ++r) {
            int h = t * 16 + r + mofs;
            if (h < HH) {
                float x = acc[t][r] + bias[h];
                x = eluf(x);
                x = fmaf(x, sc[h], sh[h]);
                x = eluf(x);
                Bmap[(size_t)h * PIX + pb] = x;
            }
        }
    }
}

// ---------------------------------------------------------------------------
// Kernel 3: C = bn2(conv7x7(Bmap, W2) + b2), padding 3, NCHW/OIHW, O=1
// ---------------------------------------------------------------------------
__global__ __launch_bounds__(256) void conv_kernel(
    const float* __restrict__ Bmap, const float* __restrict__ W2,
    const float* __restrict__ b2,
    const float* __restrict__ g2, const float* __restrict__ bb2,
    const float* __restrict__ m2, const float* __restrict__ v2,
    float* __restrict__ C)
{
    __shared__ float wsm[HH * 49];
    const int tid = threadIdx.x;
    for (int idx = tid; idx < HH * 49; idx += 256) wsm[idx] = W2[idx];
    __syncthreads();

    const int ox = blockIdx.x * 16 + (tid & 15);
    const int oy = blockIdx.y * 16 + (tid >> 4);
    float acc = 0.0f;
    for (int c = 0; c < HH; ++c) {
        const float* bp = Bmap + (size_t)c * PIX;
        const float* wc = wsm + c * 49;
        #pragma unroll
        for (int dy = 0; dy < 7; ++dy) {
            int yy = oy + dy - 3;
            if ((unsigned)yy < (unsigned)L) {
                const float* row = bp + (size_t)yy * L;
                #pragma unroll
                for (int dx = 0; dx < 7; ++dx) {
                    int xx = ox + dx - 3;
                    if ((unsigned)xx < (unsigned)L)
                        acc = fmaf(row[xx], wc[dy * 7 + dx], acc);
                }
            }
        }
    }
    float s2 = g2[0] / sqrtf(v2[0] + 1e-5f);
    C[(size_t)oy * L + ox] = fmaf(acc + b2[0] - m2[0], s2, bb2[0]);
}

// ---------------------------------------------------------------------------
// Kernel 4: 9x9 max pool, stride 9, pad 4 -> 89x89
// ---------------------------------------------------------------------------
__global__ __launch_bounds__(256) void pool_kernel(
    const float* __restrict__ C, float* __restrict__ yhat)
{
    int idx = blockIdx.x * 256 + threadIdx.x;
    if (idx >= PY * PY) return;
    int py = idx / PY, px = idx % PY;
    int y0 = py * 9 - 4, x0 = px * 9 - 4;
    float mx = -INFINITY;
    #pragma unroll
    for (int dy = 0; dy < 9; ++dy) {
        int yy = y0 + dy;
        if ((unsigned)yy >= (unsigned)L) continue;
        const float* row = C + (size_t)yy * L;
        #pragma unroll
        for (int dx = 0; dx < 9; ++dx) {
            int xx = x0 + dx;
            if ((unsigned)xx < (unsigned)L) mx = fmaxf(mx, row[xx]);
        }
    }
    yhat[idx] = mx;
}

// ---------------------------------------------------------------------------
// Kernel 5: scalar statistics + exact GELU (single block)
// ---------------------------------------------------------------------------
__global__ __launch_bounds__(1024) void final_kernel(
    const float* __restrict__ yhat, const float* __restrict__ gamma,
    float* __restrict__ out)
{
    const int N = PY * PY;
    __shared__ float r0[32], r1[32];
    __shared__ float s_thr;
    const int tid = threadIdx.x, wave = tid >> 5, lane = tid & 31;

    float s = 0.0f, s2 = 0.0f;
    for (int i = tid; i < N; i += 1024) {
        float y = yhat[i];
        s += y; s2 = fmaf(y, y, s2);
    }
    for (int off = 16; off; off >>= 1) {
        s  += __shfl_down(s, off, 32);
        s2 += __shfl_down(s2, off, 32);
    }
    if (lane == 0) { r0[wave] = s; r1[wave] = s2; }
    __syncthreads();
    if (tid == 0) {
        float ts = 0.0f, ts2 = 0.0f;
        for (int w = 0; w < 32; ++w) { ts += r0[w]; ts2 += r1[w]; }
        float mu  = ts / (float)N;
        float var = (ts2 - (float)N * mu * mu) / (float)(N - 1);  // ddof=1
        s_thr = mu + gamma[0] * var;
    }
    __syncthreads();
    float thr = s_thr;
    float qs = 0.0f, qc = 0.0f;
    for (int i = tid; i < N; i += 1024) {
        float q = yhat[i] - thr;
        if (q > 0.0f) { qs += q; qc += 1.0f; }
    }
    for (int off = 16; off; off >>= 1) {
        qs += __shfl_down(qs, off, 32);
        qc += __shfl_down(qc, off, 32);
    }
    if (lane == 0) { r0[wave] = qs; r1[wave] = qc; }
    __syncthreads();
    if (tid == 0) {
        float tq = 0.0f, tc = 0.0f;
        for (int w = 0; w < 32; ++w) { tq += r0[w]; tc += r1[w]; }
        float phat = tq / (tc + 1.0f);
        out[0] = 0.5f * phat * (1.0f + erff(phat * 0.70710678f));  // exact gelu
    }
}

// ---------------------------------------------------------------------------
extern "C" void kernel_launch(void* const* d_in, const int* in_sizes, int n_in,
                              void* d_out, int out_size, void* d_ws, size_t ws_size,
                              hipStream_t stream)
{
    const float* z0    = (const float*)d_in[0];
    const float* z1    = (const float*)d_in[1];
    const float* W_emb = (const float*)d_in[2];
    const float* b_emb = (const float*)d_in[3];
    const float* W1    = (const float*)d_in[4];
    const float* b1    = (const float*)d_in[5];
    const float* bn1_g = (const float*)d_in[6];
    const float* bn1_b = (const float*)d_in[7];
    const float* bn1_m = (const float*)d_in[8];
    const float* bn1_v = (const float*)d_in[9];
    const float* W2    = (const float*)d_in[10];
    const float* b2    = (const float*)d_in[11];
    const float* bn2_g = (const float*)d_in[12];
    const float* bn2_b = (const float*)d_in[13];
    const float* bn2_m = (const float*)d_in[14];
    const float* bn2_v = (const float*)d_in[15];
    const float* gamma = (const float*)d_in[16];

    float* ws   = (float*)d_ws;
    float* e0   = ws;                       // 80,000
    float* e1   = ws + 80000;               // 80,000
    float* Bmap = ws + 160000;              // 50*800*800 = 32,000,000
    float* C    = Bmap + (size_t)HH * PIX;  // 640,000
    float* yhat = C + PIX;                  // 7,921

    emb_kernel<<<dim3((L * DD + 255) / 256, 2), 256, 0, stream>>>(
        z0, z1, W_emb, b_emb, e0, e1);

    pair_gemm_kernel<<<dim3(L / 16, L / 16), 512, 0, stream>>>(
        e0, e1, W1, b1, bn1_g, bn1_b, bn1_m, bn1_v, Bmap);

    conv_kernel<<<dim3(L / 16, L / 16), 256, 0, stream>>>(
        Bmap, W2, b2, bn2_g, bn2_b, bn2_m, bn2_v, C);

    pool_kernel<<<(PY * PY + 255) / 256, 256, 0, stream>>>(C, yhat);

    final_kernel<<<1, 1024, 0, stream>>>(yhat, gamma, (float*)d_out);
}